// USPBlock_Inductive_MSconv_21010980012162
// MI455X (gfx1250) — compile-verified
//
#include <hip/hip_runtime.h>
#include <math.h>

// Problem constants (match reference)
#define BB 8
#define TT 12
#define NN 1000
#define CC 32
#define FF 64
#define MTILES 63    // ceil(1000/16)
#define KC 40        // K-chunk staged in LDS (1000 = 25 * 40)
#define KCP 44       // padded LDS row stride in floats -> conflict-free ds_load_b64
#define NCHUNK 25

typedef float v2f __attribute__((ext_vector_type(2)));
typedef float v4f __attribute__((ext_vector_type(4)));
typedef float v8f __attribute__((ext_vector_type(8)));

// ---------------------------------------------------------------------------
// support = Xin @ W[t] for both gate weights (dual GEMM sharing the A operand).
// Xin: (B*T, N, Cin) row-major; W: (T, Cin, 64) row-major.
// Output is written TRANSPOSED: S_T (B*T, 64, N) so the big propagation GEMM
// can fetch its B fragments as contiguous 8-byte reads.
// fp32 WMMA fragment layouts per CDNA5 ISA:
//   A 16x4 : lane m=l&15 holds row m; VGPR0=K(2h), VGPR1=K(2h+1)
//   B 4x16 : lane n=l&15 holds col n; VGPR0=K(2h), VGPR1=K(2h+1)
//   C/D    : VGPR v holds (row = v + 8h, col = l&15)
// ---------------------------------------------------------------------------
__global__ void __launch_bounds__(128) support_kernel(
    const float* __restrict__ Xin, const float* __restrict__ Wa,
    const float* __restrict__ Wb, float* __restrict__ SaT,
    float* __restrict__ SbT, int Cin)
{
  const int lane = threadIdx.x & 31;
  const int wv   = threadIdx.x >> 5;
  const int mt   = blockIdx.x * 4 + wv;
  if (mt >= MTILES) return;               // wave-uniform exit (no barriers here)
  const int bt = blockIdx.y;
  const int t  = bt % TT;

  const float* P  = Xin + (size_t)bt * NN * Cin;
  const float* Qa = Wa + (size_t)t * Cin * FF;
  const float* Qb = Wb + (size_t)t * Cin * FF;

  const int m = lane & 15;
  const int h = lane >> 4;
  const int row  = mt * 16 + m;
  const int rowc = (row < NN) ? row : (NN - 1);   // clamp: garbage rows never stored
  const float* prow = P + (size_t)rowc * Cin;

  const v8f zero = {0.f, 0.f, 0.f, 0.f, 0.f, 0.f, 0.f, 0.f};
  v8f ca[4], cb[4];
#pragma unroll
  for (int j = 0; j < 4; ++j) { ca[j] = zero; cb[j] = zero; }

  for (int k0 = 0; k0 < Cin; k0 += 4) {
    const v2f a = *(const v2f*)(prow + k0 + 2 * h);
    const float* qa0 = Qa + (size_t)(k0 + 2 * h) * FF + m;
    const float* qb0 = Qb + (size_t)(k0 + 2 * h) * FF + m;
#pragma unroll
    for (int j = 0; j < 4; ++j) {
      v2f bva, bvb;
      bva.x = qa0[16 * j];
      bva.y = qa0[FF + 16 * j];
      bvb.x = qb0[16 * j];
      bvb.y = qb0[FF + 16 * j];
      ca[j] = __builtin_amdgcn_wmma_f32_16x16x4_f32(
          false, a, false, bva, (short)0, ca[j], false, false);
      cb[j] = __builtin_amdgcn_wmma_f32_16x16x4_f32(
          false, a, false, bvb, (short)0, cb[j], false, false);
    }
  }

#pragma unroll
  for (int j = 0; j < 4; ++j) {
#pragma unroll
    for (int v = 0; v < 8; ++v) {
      const int r = mt * 16 + v + 8 * h;
      if (r < NN) {
        const size_t o = ((size_t)bt * FF + 16 * j + m) * NN + r;  // transposed
        SaT[o] = ca[j][v];
        SbT[o] = cb[j][v];
      }
    }
  }
}

// ---------------------------------------------------------------------------
// Async staging of one K-chunk of both gate operands into LDS
// (GLOBAL_LOAD_ASYNC_TO_LDS_B128, tracked by ASYNCcnt; 1 instruction / 16B,
// no VGPR round-trip). LDS byte address = low 32 bits of the flat pointer
// (ISA 10.2: LDS aperture maps addr[31:0] directly to the LDS offset).
// Layout in LDS: [f][k] with padded row stride KCP=44 -> conflict-free reads.
// ---------------------------------------------------------------------------
__device__ __forceinline__ void stage_chunk_async(
    float* dA, float* dB,
    const float* __restrict__ Qa, const float* __restrict__ Qb,
    int kb, int tid)
{
#pragma unroll
  for (int i = 0; i < 5; ++i) {
    const int e  = (i * 128 + tid) * 4;   // 0..2556
    const int f  = e / KC;
    const int kk = e - f * KC;
    const unsigned la = (unsigned)(uintptr_t)(dA + f * KCP + kk);
    const unsigned lb = (unsigned)(uintptr_t)(dB + f * KCP + kk);
    const float* ga = Qa + (size_t)f * NN + kb + kk;
    const float* gb = Qb + (size_t)f * NN + kb + kk;
    asm volatile("global_load_async_to_lds_b128 %0, %1, off"
                 :: "v"(la), "v"(ga) : "memory");
    asm volatile("global_load_async_to_lds_b128 %0, %1, off"
                 :: "v"(lb), "v"(gb) : "memory");
  }
}

__device__ __forceinline__ void wait_async()
{
  asm volatile("s_wait_asynccnt 0" ::: "memory");
}

// Epilogue: bias + gate, then mode-dependent combine/store.
__device__ __forceinline__ void gated_epilogue(
    int mt, int m, int h, int t, int b, int bt, int mode,
    const float* __restrict__ Ba, const float* __restrict__ Bb,
    float* __restrict__ Hout, const float* __restrict__ Hprev,
    const float* __restrict__ Xres, const v8f ca[4], const v8f cb[4])
{
#pragma unroll
  for (int j = 0; j < 4; ++j) {
    const int f = 16 * j + m;
    const float biasA = Ba[t * FF + f];
    const float biasB = Bb[t * FF + f];
#pragma unroll
    for (int v = 0; v < 8; ++v) {
      const int r = mt * 16 + v + 8 * h;
      if (r < NN) {
        const float ya = ca[j][v] + biasA;
        const float yb = cb[j][v] + biasB;
        const float hv = ya * (1.0f / (1.0f + __expf(-yb)));
        const size_t o = ((size_t)bt * NN + r) * FF + f;   // row-major output
        if (mode == 0) {
          Hout[o] = hv;
        } else if (mode == 1) {
          Hout[o] = fmaxf(Hprev[o], hv);
        } else {
          const float blkT = fmaxf(Hprev[o], hv);
          const float xr = Xres[(((size_t)b * NN + r) * FF + f) * TT + t];
          Hout[o] = fmaxf(Hout[o], blkT) + xr;
        }
      }
    }
  }
}

// ---------------------------------------------------------------------------
// Gated graph propagation: H = (A@Sa + ba) * sigmoid(A@Sb + bb)
//   mode 0: Hout = H
//   mode 1: Hout = max(Hprev, H)
//   mode 2: Hout = max(Hout, max(Hprev, H)) + Xres
// Each wave accumulates TWO 16-row tiles (mt, mt+32): every LDS B-fragment
// feeds 2 WMMAs -> per K-step: 2 global b64 + 8 ds_load_b64 + 16 WMMA.
// The 4 waves of a block share the LDS-staged S operands (async copies).
// ---------------------------------------------------------------------------
__global__ void __launch_bounds__(128) gated_gcn_kernel(
    const float* __restrict__ A, const float* __restrict__ SaT,
    const float* __restrict__ SbT, const float* __restrict__ Ba,
    const float* __restrict__ Bb, float* __restrict__ Hout,
    const float* __restrict__ Hprev, const float* __restrict__ Xres, int mode)
{
  __shared__ float lds_s[2][2][FF * KCP];   // ~45 KB

  const int tid  = threadIdx.x;
  const int lane = tid & 31;
  const int wv   = tid >> 5;
  const int mt0  = blockIdx.x * 4 + wv;     // 0..31
  const int mt1  = mt0 + 32;                // 32..63 (tile 63 = pad)
  const int bt   = blockIdx.y;
  const int t    = bt % TT;
  const int b    = bt / TT;

  const float* Qa = SaT + (size_t)bt * FF * NN;
  const float* Qb = SbT + (size_t)bt * FF * NN;

  const int m = lane & 15;
  const int h = lane >> 4;
  const int row0 = mt0 * 16 + m;
  const int row1 = mt1 * 16 + m;
  const int r0c = (row0 < NN) ? row0 : (NN - 1);  // clamp (pad rows never stored)
  const int r1c = (row1 < NN) ? row1 : (NN - 1);
  const float* prow0 = A + (size_t)r0c * NN;
  const float* prow1 = A + (size_t)r1c * NN;

  const v8f zero = {0.f, 0.f, 0.f, 0.f, 0.f, 0.f, 0.f, 0.f};
  v8f ca0[4], cb0[4], ca1[4], cb1[4];
#pragma unroll
  for (int j = 0; j < 4; ++j) { ca0[j] = zero; cb0[j] = zero; ca1[j] = zero; cb1[j] = zero; }

  // prime the pipeline: stage chunk 0 into buffer 0
  stage_chunk_async(&lds_s[0][0][0], &lds_s[0][1][0], Qa, Qb, 0, tid);
  wait_async();
  __syncthreads();

  for (int c = 0; c < NCHUNK; ++c) {
    const int cur = c & 1;
    if (c + 1 < NCHUNK)
      stage_chunk_async(&lds_s[cur ^ 1][0][0], &lds_s[cur ^ 1][1][0],
                        Qa, Qb, (c + 1) * KC, tid);

    // prefetch next A chunk into cache (global_prefetch_b8)
    __builtin_prefetch(prow0 + c * KC + KC + 2 * h, 0, 0);
    __builtin_prefetch(prow1 + c * KC + KC + 2 * h, 0, 0);

    const float* ldsA = &lds_s[cur][0][0];
    const float* ldsB = &lds_s[cur][1][0];
#pragma unroll
    for (int kk0 = 0; kk0 < KC; kk0 += 4) {
      const v2f a0 = *(const v2f*)(prow0 + c * KC + kk0 + 2 * h);
      const v2f a1 = *(const v2f*)(prow1 + c * KC + kk0 + 2 * h);
#pragma unroll
      for (int j = 0; j < 4; ++j) {
        const v2f bva = *(const v2f*)(ldsA + (16 * j + m) * KCP + kk0 + 2 * h);
        const v2f bvb = *(const v2f*)(ldsB + (16 * j + m) * KCP + kk0 + 2 * h);
        ca0[j] = __builtin_amdgcn_wmma_f32_16x16x4_f32(
            false, a0, false, bva, (short)0, ca0[j], false, false);
        cb0[j] = __builtin_amdgcn_wmma_f32_16x16x4_f32(
            false, a0, false, bvb, (short)0, cb0[j], false, false);
        ca1[j] = __builtin_amdgcn_wmma_f32_16x16x4_f32(
            false, a1, false, bva, (short)0, ca1[j], false, false);
        cb1[j] = __builtin_amdgcn_wmma_f32_16x16x4_f32(
            false, a1, false, bvb, (short)0, cb1[j], false, false);
      }
    }
    wait_async();      // this wave's async fills complete before the barrier
    __syncthreads();   // all fills visible; buffer 'cur' free for refill
  }

  gated_epilogue(mt0, m, h, t, b, bt, mode, Ba, Bb, Hout, Hprev, Xres, ca0, cb0);
  gated_epilogue(mt1, m, h, t, b, bt, mode, Ba, Bb, Hout, Hprev, Xres, ca1, cb1);
}

// ---------------------------------------------------------------------------
// Dilated causal TCN layer (K=2 taps: w0*x[t-d] + w1*x[t], bias, ReLU).
// firstLayer: input is X (B,T,N,Ci); otherwise (B,N,Ci,T). Output (B,N,Co,T).
// Negligible FLOPs -> plain VALU.
// ---------------------------------------------------------------------------
__global__ void tcn_conv_kernel(
    const float* __restrict__ in, const float* __restrict__ w,
    const float* __restrict__ bias, float* __restrict__ out,
    int Ci, int Co, int dil, int firstLayer)
{
  const int idx = blockIdx.x * blockDim.x + threadIdx.x;
  const int total = BB * NN * Co;
  if (idx >= total) return;
  const int co = idx % Co;
  const int n  = (idx / Co) % NN;
  const int b  = idx / (Co * NN);

  float acc[TT];
  const float bv = bias[co];
#pragma unroll
  for (int t = 0; t < TT; ++t) acc[t] = bv;

  for (int ci = 0; ci < Ci; ++ci) {
    const float w0 = w[(co * Ci + ci) * 2 + 0];
    const float w1 = w[(co * Ci + ci) * 2 + 1];
    float xv[TT];
    if (firstLayer) {
#pragma unroll
      for (int t = 0; t < TT; ++t)
        xv[t] = in[(((size_t)b * TT + t) * NN + n) * Ci + ci];
    } else {
#pragma unroll
      for (int t = 0; t < TT; ++t)
        xv[t] = in[(((size_t)b * NN + n) * Ci + ci) * TT + t];
    }
#pragma unroll
    for (int t = 0; t < TT; ++t) {
      const float x0 = (t >= dil) ? xv[t - dil] : 0.f;
      acc[t] += w0 * x0 + w1 * xv[t];
    }
  }
#pragma unroll
  for (int t = 0; t < TT; ++t)
    out[(((size_t)b * NN + n) * Co + co) * TT + t] = fmaxf(acc[t], 0.f);
}

// ---------------------------------------------------------------------------
extern "C" void kernel_launch(void* const* d_in, const int* in_sizes, int n_in,
                              void* d_out, int out_size, void* d_ws,
                              size_t ws_size, hipStream_t stream)
{
  const float* X   = (const float*)d_in[0];
  const float* A_s = (const float*)d_in[1];
  const float* A_t = (const float*)d_in[2];
  const float* cw0 = (const float*)d_in[3];
  const float* cb0 = (const float*)d_in[4];
  const float* cw1 = (const float*)d_in[5];
  const float* cb1 = (const float*)d_in[6];
  const float* cw2 = (const float*)d_in[7];
  const float* cb2 = (const float*)d_in[8];
  const float* ts[8];
  const float* tp[8];
  for (int i = 0; i < 8; ++i) ts[i] = (const float*)d_in[9 + i];
  for (int i = 0; i < 8; ++i) tp[i] = (const float*)d_in[17 + i];
  // ts/tp order: w1a, b1a, w1b, b1b, w2a, b2a, w2b, b2b

  float* out = (float*)d_out;
  float* ws  = (float*)d_ws;
  const size_t SL = (size_t)BB * TT * NN * FF;  // 6,144,000 floats per slab
  float* xres = ws;            // (B,N,64,T)
  float* bufA = ws + SL;       // SaT slab / TCN intermediate 0
  float* bufB = ws + 2 * SL;   // SbT slab
  float* h1   = ws + 3 * SL;   // h1 slab (row-major) / TCN intermediate 1

  const dim3 gs(16, BB * TT, 1);   // support: 64 waves of 1 tile (63 + pad)
  const dim3 gg(8,  BB * TT, 1);   // gated:   32 waves of 2 tiles (63 + pad)
  const dim3 bw(128, 1, 1);

  // --- TCN residual branch (x_res) ---
  {
    int th0 = BB * NN * 64;
    tcn_conv_kernel<<<(th0 + 255) / 256, 256, 0, stream>>>(X, cw0, cb0, bufA, 32, 64, 1, 1);
    int th1 = BB * NN * 32;
    tcn_conv_kernel<<<(th1 + 255) / 256, 256, 0, stream>>>(bufA, cw1, cb1, h1, 64, 32, 2, 0);
    int th2 = BB * NN * 64;
    tcn_conv_kernel<<<(th2 + 255) / 256, 256, 0, stream>>>(h1, cw2, cb2, xres, 32, 64, 4, 0);
  }

  // --- Spatial block (A_s); result staged in d_out ---
  support_kernel<<<gs, bw, 0, stream>>>(X, ts[0], ts[2], bufA, bufB, CC);
  gated_gcn_kernel<<<gg, bw, 0, stream>>>(A_s, bufA, bufB, ts[1], ts[3], h1, nullptr, nullptr, 0);
  support_kernel<<<gs, bw, 0, stream>>>(h1, ts[4], ts[6], bufA, bufB, FF);
  gated_gcn_kernel<<<gg, bw, 0, stream>>>(A_s, bufA, bufB, ts[5], ts[7], out, h1, nullptr, 1);

  // --- Temporal block (A_t); fused final max + x_res ---
  support_kernel<<<gs, bw, 0, stream>>>(X, tp[0], tp[2], bufA, bufB, CC);
  gated_gcn_kernel<<<gg, bw, 0, stream>>>(A_t, bufA, bufB, tp[1], tp[3], h1, nullptr, nullptr, 0);
  support_kernel<<<gs, bw, 0, stream>>>(h1, tp[4], tp[6], bufA, bufB, FF);
  gated_gcn_kernel<<<gg, bw, 0, stream>>>(A_t, bufA, bufB, tp[5], tp[7], out, h1, xres, 2);
}